// PersImClassifier_43482248905288
// MI455X (gfx1250) — compile-verified
//
#include <hip/hip_runtime.h>
#include <hip/hip_bf16.h>
#include <math.h>

typedef __attribute__((ext_vector_type(2))) float v2f;
typedef __attribute__((ext_vector_type(8))) float v8f;

// ---------------------------------------------------------------------------
// Stage A: persistence images as separable-KDE GEMM.
// One block per (b, c) image (64 blocks).  Per 128-point K-chunk:
//   Ex[i][k] = exp(-50*(c0[k]-0.01*i)^2), Ey[j][k] = exp(-50*(c1[k]-0.01*j)^2)
// then im[i][j] = sum_k Ex[i][k]*Ey[j][k] accumulated with f32 WMMA 16x16x4.
// 7x7 tiles of 16x16 across 8 waves; then block-max reduce + normalized store.
// ---------------------------------------------------------------------------
#define A_STRIDE 132   // 128 K + pad (132 % 64 = 4 -> conflict-free row access)

__global__ __launch_bounds__(256) void pers_image_kernel(
    const float* __restrict__ h0,  const float* __restrict__ h0e,
    const float* __restrict__ h1,  const float* __restrict__ h1e,
    float* __restrict__ img) {
  extern __shared__ float smem[];
  float* sEx  = smem;                    // 112 * 132
  float* sEy  = sEx + 112 * A_STRIDE;    // 112 * 132
  float* sPts = sEy + 112 * A_STRIDE;    // 256  (c0 | c1)
  float* sRed = sPts + 256;              // 256

  const int blk = blockIdx.x;
  const int b = blk >> 2, c = blk & 3;
  const float* src = (c == 0) ? h0 : (c == 1) ? h0e : (c == 2) ? h1 : h1e;

  const int tid  = threadIdx.x;
  const int wave = tid >> 5, lane = tid & 31;
  const int half = lane >> 4, lmod = lane & 15;

  v8f acc[7] = {};

  for (int kc = 0; kc < 4; ++kc) {
    // load & preprocess 128 birth/death pairs
    if (tid < 128) {
      int n = kc * 128 + tid;
      float a0 = src[(b * 512 + n) * 2 + 0];
      float a1 = src[(b * 512 + n) * 2 + 1];
      float bm = fminf(a0, a1), dm = fmaxf(a0, a1);
      sPts[tid]       = fabsf(bm);        // birth
      sPts[128 + tid] = fabsf(dm - bm);   // persistence
    }
    __syncthreads();
    // separable Gaussian factors: 1/(2*IM_VAR) = 50
    for (int idx = tid; idx < 100 * 128; idx += 256) {
      int i = idx >> 7, k = idx & 127;
      float x  = 0.01f * (float)i;
      float dx = sPts[k] - x;
      float dy = sPts[128 + k] - x;
      sEx[i * A_STRIDE + k] = __expf(-50.f * dx * dx);
      sEy[i * A_STRIDE + k] = __expf(-50.f * dy * dy);
    }
    __syncthreads();
    // rank-128 update of all 49 tiles via V_WMMA_F32_16X16X4_F32
    #pragma unroll
    for (int i = 0; i < 7; ++i) {
      int t = wave + i * 8;
      if (t < 49) {
        int m0 = (t / 7) * 16, n0 = (t % 7) * 16;
        v8f a8 = acc[i];
        for (int k = 0; k < 128; k += 4) {
          v2f av = *(const v2f*)(sEx + (m0 + lmod) * A_STRIDE + k + 2 * half);
          v2f bv = *(const v2f*)(sEy + (n0 + lmod) * A_STRIDE + k + 2 * half);
          a8 = __builtin_amdgcn_wmma_f32_16x16x4_f32(
              false, av, false, bv, (short)0, a8, false, false);
        }
        acc[i] = a8;
      }
    }
    __syncthreads();
  }

  // block-wide max over valid pixels
  float m = 0.f;
  #pragma unroll
  for (int i = 0; i < 7; ++i) {
    int t = wave + i * 8;
    if (t < 49) {
      int m0 = (t / 7) * 16, n0 = (t % 7) * 16;
      #pragma unroll
      for (int v = 0; v < 8; ++v) {
        int ii = m0 + v + 8 * half, jj = n0 + lmod;
        if (ii < 100 && jj < 100) m = fmaxf(m, acc[i][v]);
      }
    }
  }
  sRed[tid] = m;
  __syncthreads();
  for (int s = 128; s > 0; s >>= 1) {
    if (tid < s) sRed[tid] = fmaxf(sRed[tid], sRed[tid + s]);
    __syncthreads();
  }
  // im/(max+eps) with 1/(2*pi*var) folded: S/(maxS + 2*pi*var*eps)
  float scale = 1.f / (sRed[0] + 6.2831853e-8f);

  float* dst = img + (size_t)blk * 10000;
  #pragma unroll
  for (int i = 0; i < 7; ++i) {
    int t = wave + i * 8;
    if (t < 49) {
      int m0 = (t / 7) * 16, n0 = (t % 7) * 16;
      #pragma unroll
      for (int v = 0; v < 8; ++v) {
        int ii = m0 + v + 8 * half, jj = n0 + lmod;
        if (ii < 100 && jj < 100) dst[ii * 100 + jj] = acc[i][v] * scale;
      }
    }
  }
}

// ---------------------------------------------------------------------------
// Stage B1: conv1 (4->16ch, 5x5, pad 2) + bias + ReLU as implicit GEMM.
// One block per (b, output row).  M=16 (Cout, one tile), K=100, N=100 px.
// ---------------------------------------------------------------------------
__global__ __launch_bounds__(256) void conv1_kernel(
    const float* __restrict__ img, const float* __restrict__ w,
    const float* __restrict__ bias, float* __restrict__ out) {
  extern __shared__ float smem[];
  float* sW   = smem;          // 16 * 100
  float* sCol = sW + 1600;     // 112 * 104 (im2col, [px][k])

  const int blk = blockIdx.x;
  const int b = blk / 100, oy = blk % 100;
  const int tid = threadIdx.x;

  for (int idx = tid; idx < 1600; idx += 256) sW[idx] = w[idx];
  const float* imb = img + (size_t)b * 4 * 10000;
  for (int idx = tid; idx < 100 * 100; idx += 256) {
    int ox = idx / 100, k = idx % 100;
    int cin = k / 25, r = k % 25, kh = r / 5, kw = r % 5;
    int iy = oy + kh - 2, ix = ox + kw - 2;
    float v = 0.f;
    if (iy >= 0 && iy < 100 && ix >= 0 && ix < 100)
      v = imb[cin * 10000 + iy * 100 + ix];
    sCol[ox * 104 + k] = v;
  }
  __syncthreads();

  const int wave = tid >> 5, lane = tid & 31;
  const int half = lane >> 4, lmod = lane & 15;
  if (wave < 7) {
    int n0 = wave * 16;
    v8f acc = {};
    for (int k = 0; k < 100; k += 4) {
      v2f av = *(const v2f*)(sW + lmod * 100 + k + 2 * half);
      v2f bv = *(const v2f*)(sCol + (n0 + lmod) * 104 + k + 2 * half);
      acc = __builtin_amdgcn_wmma_f32_16x16x4_f32(
          false, av, false, bv, (short)0, acc, false, false);
    }
    #pragma unroll
    for (int v = 0; v < 8; ++v) {
      int ch = v + 8 * half, ox = n0 + lmod;
      if (ox < 100)
        out[((size_t)(b * 16 + ch) * 100 + oy) * 100 + ox] =
            fmaxf(acc[v] + bias[ch], 0.f);
    }
  }
}

// ---------------------------------------------------------------------------
// 2x2 max-pool (stride 2), elementwise.
// ---------------------------------------------------------------------------
__global__ void pool2x2_kernel(const float* __restrict__ in,
                               float* __restrict__ out,
                               int total, int H, int W) {
  int idx = blockIdx.x * blockDim.x + threadIdx.x;
  if (idx >= total) return;
  int px = idx % W;
  int t = idx / W;
  int py = t % H;
  int bc = t / H;
  int inW = W * 2;
  const float* p = in + ((size_t)bc * H * 2 + py * 2) * inW + px * 2;
  out[idx] = fmaxf(fmaxf(p[0], p[1]), fmaxf(p[inW], p[inW + 1]));
}

// ---------------------------------------------------------------------------
// Stage C1: conv2 (16->32ch, 5x5, pad 2) + bias + ReLU, implicit GEMM.
// One block per (b, output row).  M=32 (2 tiles), K=400, N=50 px (4 tiles);
// 8 tile-jobs -> exactly one per wave.
// ---------------------------------------------------------------------------
#define W2_STRIDE 404  // 400 K + pad (404 % 64 = 20 -> conflict-free)

__global__ __launch_bounds__(256) void conv2_kernel(
    const float* __restrict__ pin, const float* __restrict__ w,
    const float* __restrict__ bias, float* __restrict__ out) {
  extern __shared__ float smem[];
  float* sW   = smem;                  // 32 * 404
  float* sCol = sW + 32 * W2_STRIDE;   // 64 * 404

  const int blk = blockIdx.x;
  const int b = blk / 50, oy = blk % 50;
  const int tid = threadIdx.x;

  for (int idx = tid; idx < 32 * 400; idx += 256) {
    int m = idx / 400, k = idx % 400;
    sW[m * W2_STRIDE + k] = w[idx];
  }
  const float* pb = pin + (size_t)b * 16 * 2500;
  for (int idx = tid; idx < 50 * 400; idx += 256) {
    int ox = idx / 400, k = idx % 400;
    int cin = k / 25, r = k % 25, kh = r / 5, kw = r % 5;
    int iy = oy + kh - 2, ix = ox + kw - 2;
    float v = 0.f;
    if (iy >= 0 && iy < 50 && ix >= 0 && ix < 50)
      v = pb[cin * 2500 + iy * 50 + ix];
    sCol[ox * W2_STRIDE + k] = v;
  }
  __syncthreads();

  const int wave = tid >> 5, lane = tid & 31;
  const int half = lane >> 4, lmod = lane & 15;
  const int m0 = (wave >> 2) * 16, n0 = (wave & 3) * 16;
  v8f acc = {};
  for (int k = 0; k < 400; k += 4) {
    v2f av = *(const v2f*)(sW + (m0 + lmod) * W2_STRIDE + k + 2 * half);
    v2f bv = *(const v2f*)(sCol + (n0 + lmod) * W2_STRIDE + k + 2 * half);
    acc = __builtin_amdgcn_wmma_f32_16x16x4_f32(
        false, av, false, bv, (short)0, acc, false, false);
  }
  #pragma unroll
  for (int v = 0; v < 8; ++v) {
    int ch = m0 + v + 8 * half, ox = n0 + lmod;
    if (ox < 50)
      out[((size_t)(b * 32 + ch) * 50 + oy) * 50 + ox] =
          fmaxf(acc[v] + bias[ch], 0.f);
  }
}

// ---------------------------------------------------------------------------
// Stage D: FC  out[16,10] = X[16,20000] @ W^T + b.
// Single block: M=16 (batch), N=16 (10 classes padded), split-K over 8 waves
// (2500 each), A/B fragments loaded straight from global as float2, partials
// reduced in LDS.
// ---------------------------------------------------------------------------
__global__ __launch_bounds__(256) void fc_kernel(
    const float* __restrict__ x, const float* __restrict__ w,
    const float* __restrict__ bias, float* __restrict__ out) {
  __shared__ float sAcc[8 * 256];
  const int tid = threadIdx.x;
  const int wave = tid >> 5, lane = tid & 31;
  const int half = lane >> 4, lmod = lane & 15;

  const float wm = (lmod < 10) ? 1.f : 0.f;
  const float* xr = x + (size_t)lmod * 20000;
  const float* wr = w + (size_t)((lmod < 10) ? lmod : 0) * 20000;

  const int k0 = wave * 2500;
  v8f acc = {};
  for (int k = k0; k < k0 + 2500; k += 4) {
    v2f av = *(const v2f*)(xr + k + 2 * half);
    v2f bv = *(const v2f*)(wr + k + 2 * half) * wm;  // zero pad rows 10..15
    acc = __builtin_amdgcn_wmma_f32_16x16x4_f32(
        false, av, false, bv, (short)0, acc, false, false);
  }
  #pragma unroll
  for (int v = 0; v < 8; ++v) sAcc[wave * 256 + lane * 8 + v] = acc[v];
  __syncthreads();

  float tot = 0.f;
  #pragma unroll
  for (int w8 = 0; w8 < 8; ++w8) tot += sAcc[w8 * 256 + tid];
  int l = tid >> 3, v = tid & 7;
  int m = v + 8 * (l >> 4), n = l & 15;
  if (n < 10) out[m * 10 + n] = tot + bias[n];
}

// ---------------------------------------------------------------------------
extern "C" void kernel_launch(void* const* d_in, const int* in_sizes, int n_in,
                              void* d_out, int out_size, void* d_ws, size_t ws_size,
                              hipStream_t stream) {
  const float* h0      = (const float*)d_in[0];
  const float* h0e     = (const float*)d_in[1];
  const float* h1      = (const float*)d_in[2];
  const float* h1e     = (const float*)d_in[3];
  const float* conv1_w = (const float*)d_in[4];
  const float* conv1_b = (const float*)d_in[5];
  const float* conv2_w = (const float*)d_in[6];
  const float* conv2_b = (const float*)d_in[7];
  const float* fc_w    = (const float*)d_in[8];
  const float* fc_b    = (const float*)d_in[9];

  float* ws  = (float*)d_ws;
  float* IMG = ws;                 // 16*4*100*100   =   640,000 f
  float* C1  = IMG + 640000;       // 16*16*100*100  = 2,560,000 f
  float* P1  = C1 + 2560000;       // 16*16*50*50    =   640,000 f
  float* C2  = P1 + 640000;        // 16*32*50*50    = 1,280,000 f
  float* P2  = C2 + 1280000;       // 16*32*25*25    =   320,000 f

  const int shA  = (2 * 112 * A_STRIDE + 512) * 4;            // 120,320 B
  const int shB1 = (1600 + 112 * 104) * 4;                    //  52,992 B
  const int shC1 = ((32 + 64) * W2_STRIDE) * 4;               // 155,136 B

  hipFuncSetAttribute((const void*)pers_image_kernel,
                      hipFuncAttributeMaxDynamicSharedMemorySize, shA);
  hipFuncSetAttribute((const void*)conv2_kernel,
                      hipFuncAttributeMaxDynamicSharedMemorySize, shC1);

  pers_image_kernel<<<64, 256, shA, stream>>>(h0, h0e, h1, h1e, IMG);
  conv1_kernel<<<16 * 100, 256, shB1, stream>>>(IMG, conv1_w, conv1_b, C1);
  pool2x2_kernel<<<(640000 + 255) / 256, 256, 0, stream>>>(C1, P1, 640000, 50, 50);
  conv2_kernel<<<16 * 50, 256, shC1, stream>>>(P1, conv2_w, conv2_b, C2);
  pool2x2_kernel<<<(320000 + 255) / 256, 256, 0, stream>>>(C2, P2, 320000, 25, 25);
  fc_kernel<<<1, 256, 0, stream>>>(P2, fc_w, fc_b, (float*)d_out);
}